// ExpandMultiheadAttention_3925600108643
// MI455X (gfx1250) — compile-verified
//
#include <hip/hip_runtime.h>
#include <cstdint>
#include <cstddef>

// ---------------------------------------------------------------------------
// Problem constants (from reference)
// ---------------------------------------------------------------------------
#define EMBED   1024
#define NHEAD   16
#define HDIM    64          // EMBED / NHEAD
#define BATCH   2
#define SEQLEN  1024        // S
#define TLEN    1024        // T
#define NGRP    4           // G
#define SCALE_Q 0.125f      // HDIM^-0.5
#define PITCH   68          // LDS row pitch (floats): 8B-aligned, bank-clean

typedef __attribute__((ext_vector_type(2))) float v2f;
typedef __attribute__((ext_vector_type(8))) float v8f;

static __device__ __forceinline__ v8f zero8() {
  v8f z;
#pragma unroll
  for (int i = 0; i < 8; ++i) z[i] = 0.0f;
  return z;
}

// D = A(16x4 f32) * B(4x16 f32) + C, full fp32 — matches reference numerics.
static __device__ __forceinline__ v8f wmma_f32(v2f a, v2f b, v8f c) {
  return __builtin_amdgcn_wmma_f32_16x16x4_f32(
      /*neg_a=*/false, a, /*neg_b=*/false, b,
      /*c_mod=*/(short)0, c, /*reuse_a=*/false, /*reuse_b=*/false);
}

// CDNA5 async global->LDS copy (tracked by ASYNCcnt). lds_off is the LDS byte
// address: low 32 bits of a generic pointer to __shared__ (ISA aperture rule).
static __device__ __forceinline__ void async_b128(uint32_t lds_off,
                                                  const void* gaddr) {
  asm volatile("global_load_async_to_lds_b128 %0, %1, off"
               :
               : "v"(lds_off), "v"(gaddr)
               : "memory");
}
static __device__ __forceinline__ void wait_async8() {
  asm volatile("s_wait_asynccnt 0x8" ::: "memory");
}
static __device__ __forceinline__ void wait_async0() {
  asm volatile("s_wait_asynccnt 0x0" ::: "memory");
}

// Cooperative stage of a 64-row x 64-float tile into an LDS tile of row pitch
// PITCH floats. 128 threads x 8 async b128 ops = 64*256B.
static __device__ __forceinline__ void stage_tile(uint32_t lds_base,
                                                  const float* __restrict__ g,
                                                  int gstride) {
  const int tid = threadIdx.x;
#pragma unroll
  for (int seg = 0; seg < 8; ++seg) {
    const int c   = seg * 128 + tid;   // 1024 x 16B chunks
    const int row = c >> 4;
    const int col = c & 15;
    async_b128(lds_base + (uint32_t)(row * PITCH + col * 4) * 4,
               g + (size_t)row * gstride + col * 4);
  }
}

// ---------------------------------------------------------------------------
// Projection GEMM: C = X @ W^T + bias,  X:(M,1024) f32, W:(1024,1024) f32.
// Block = 4 waves = 64x64 output tile; W tile double-buffered in LDS via
// async copies (shared by all 4 waves); A-fragments preloaded per K-chunk.
// mode 0: head-split out[(b*16+h)*L + t][hd] (x scale) | mode 2: V^T | 3: flat
// ---------------------------------------------------------------------------
__global__ void __launch_bounds__(128)
proj_kernel(const float* __restrict__ X, const float* __restrict__ W,
            const float* __restrict__ bias, float* __restrict__ out,
            int mode, float scale) {
  __shared__ float ldsW[2][64][PITCH];

  const int lane = threadIdx.x & 31;
  const int wave = threadIdx.x >> 5;
  const int half = lane >> 4;   // 0: lanes 0-15, 1: lanes 16-31
  const int l16  = lane & 15;

  const int row0 = blockIdx.x * 64 + wave * 16;  // M tile (per wave)
  const int n0   = blockIdx.y * 64;              // N tile (per block)

  const float* xrow = X + (size_t)(row0 + l16) * EMBED + half * 2;

  v8f acc[4];
#pragma unroll
  for (int nt = 0; nt < 4; ++nt) acc[nt] = zero8();

  stage_tile((uint32_t)(uintptr_t)&ldsW[0][0][0], W + (size_t)n0 * EMBED,
             EMBED);

  for (int ch = 0; ch < 16; ++ch) {               // 16 chunks of K=64
    if (ch + 1 < 16) {
      stage_tile((uint32_t)(uintptr_t)&ldsW[(ch + 1) & 1][0][0],
                 W + (size_t)n0 * EMBED + (ch + 1) * 64, EMBED);
      wait_async8();                              // chunk ch landed
    } else {
      wait_async0();
    }
    __syncthreads();                              // visible to all waves

    const float(*wb)[PITCH] = ldsW[ch & 1];

    v2f af[16];                                   // batched A-fragment clause
#pragma unroll
    for (int kcl = 0; kcl < 16; ++kcl) {
      const float* p = xrow + ch * 64 + kcl * 4;
      af[kcl][0] = p[0];
      af[kcl][1] = p[1];
    }
#pragma unroll
    for (int kcl = 0; kcl < 16; ++kcl) {
      const int kb = kcl * 4 + half * 2;
#pragma unroll
      for (int nt = 0; nt < 4; ++nt) {
        v2f b;
        b[0] = wb[nt * 16 + l16][kb];
        b[1] = wb[nt * 16 + l16][kb + 1];
        acc[nt] = wmma_f32(af[kcl], b, acc[nt]);
      }
    }
    __syncthreads();                              // done reading buf (ch&1)
  }

#pragma unroll
  for (int nt = 0; nt < 4; ++nt) {
    const int n   = n0 + nt * 16 + l16;
    const float bv = bias[n];
#pragma unroll
    for (int r = 0; r < 8; ++r) {
      const int m     = row0 + r + half * 8;      // C row = r + 8*(lane/16)
      const float val = (acc[nt][r] + bv) * scale;
      size_t idx;
      if (mode == 3) {
        idx = (size_t)m * EMBED + n;
      } else {
        const int t  = m >> 1;                    // X row = t*B + b  (B=2)
        const int bb = m & 1;
        const int h  = n >> 6;
        const int hd = n & 63;
        const int bh = bb * NHEAD + h;
        if (mode == 2) idx = ((size_t)bh * HDIM + hd) * SEQLEN + t;   // V^T
        else           idx = ((size_t)bh * SEQLEN + t) * HDIM + hd;   // Q / K
      }
      out[idx] = val;
    }
  }
}

// ---------------------------------------------------------------------------
// Fused attention per (bh, g): pass 1 = online softmax stats (QK^T streamed,
// K tile double-buffered in LDS); pass 2 = recompute scores, normalized P
// written once to d_out, transposed via LDS, P@V accumulated (K + V staged).
// ---------------------------------------------------------------------------
__global__ void __launch_bounds__(128)
attn_kernel(const float* __restrict__ q, const float* __restrict__ k,
            const float* __restrict__ vt, const float* __restrict__ gmask,
            const unsigned char* __restrict__ kpm,
            float* __restrict__ probs, float* __restrict__ attn_out) {
  __shared__ float ldsK[2][64][PITCH];
  __shared__ float ldsV[64][PITCH];
  __shared__ float pt[4][16][17];   // per-wave padded P-transpose buffer

  const int lane = threadIdx.x & 31;
  const int wave = threadIdx.x >> 5;
  const int half = lane >> 4;
  const int l16  = lane & 15;

  const int t0 = blockIdx.x * 64 + wave * 16;
  const int g  = blockIdx.y;
  const int bh = blockIdx.z;
  const int b  = bh >> 4;
  const int h  = bh & 15;

  // Q A-fragments for all 16 K-chunks of HDIM (kept in VGPRs both passes).
  v2f qf[16];
  {
    const float* qrow = q + ((size_t)bh * TLEN + (t0 + l16)) * HDIM + half * 2;
#pragma unroll
    for (int kc = 0; kc < 16; ++kc) {
      qf[kc][0] = qrow[kc * 4];
      qf[kc][1] = qrow[kc * 4 + 1];
    }
  }

  const float* kbase = k + (size_t)bh * SEQLEN * HDIM;
  const float* vtbase = vt + (size_t)bh * HDIM * SEQLEN;
  const float* mrow  = gmask + ((size_t)b * NGRP + g) * SEQLEN;
  const unsigned char* prow = kpm + (size_t)b * SEQLEN;

  float m_r[8], l_r[8];
#pragma unroll
  for (int r = 0; r < 8; ++r) { m_r[r] = -INFINITY; l_r[r] = 0.0f; }

  // ---- pass 1: online (rowmax, rowsum), K staged via async double buffer --
  stage_tile((uint32_t)(uintptr_t)&ldsK[0][0][0], kbase, HDIM);
  for (int ch = 0; ch < 16; ++ch) {               // 64 S-columns per chunk
    if (ch + 1 < 16) {
      stage_tile((uint32_t)(uintptr_t)&ldsK[(ch + 1) & 1][0][0],
                 kbase + (size_t)(ch + 1) * 64 * HDIM, HDIM);
      wait_async8();
    } else {
      wait_async0();
    }
    __syncthreads();
    const float(*kb)[PITCH] = ldsK[ch & 1];
#pragma unroll
    for (int ctl = 0; ctl < 4; ++ctl) {
      const int sl   = ctl * 16 + l16;            // local score column
      const int scol = ch * 64 + sl;
      v8f c = zero8();
#pragma unroll
      for (int kc = 0; kc < 16; ++kc) {
        const int kbi = kc * 4 + half * 2;
        v2f bf;
        bf[0] = kb[sl][kbi];
        bf[1] = kb[sl][kbi + 1];
        c = wmma_f32(qf[kc], bf, c);
      }
      const float mk = mrow[scol] + (prow[scol] ? -INFINITY : 0.0f);
#pragma unroll
      for (int r = 0; r < 8; ++r) {
        const float s  = c[r] + mk;
        const float mn = fmaxf(m_r[r], s);
        l_r[r] = l_r[r] * __expf(m_r[r] - mn) + __expf(s - mn);
        m_r[r] = mn;
      }
    }
    __syncthreads();
  }

  // Reduce (m,l) across the 16 column lanes of each half-wave group.
#pragma unroll
  for (int off = 8; off >= 1; off >>= 1) {
#pragma unroll
    for (int r = 0; r < 8; ++r) {
      const float mo = __shfl_xor(m_r[r], off, 32);
      const float lo = __shfl_xor(l_r[r], off, 32);
      const float mn = fmaxf(m_r[r], mo);
      l_r[r] = l_r[r] * __expf(m_r[r] - mn) + lo * __expf(mo - mn);
      m_r[r] = mn;
    }
  }
  float inv_l[8];
#pragma unroll
  for (int r = 0; r < 8; ++r) inv_l[r] = 1.0f / l_r[r];

  // ---- pass 2: normalized P -> d_out (write-once) + P @ V ----------------
  v8f acc[4];
#pragma unroll
  for (int nt = 0; nt < 4; ++nt) acc[nt] = zero8();

  float* pbase = probs + ((size_t)(bh * NGRP + g) * TLEN) * SEQLEN;

  stage_tile((uint32_t)(uintptr_t)&ldsK[0][0][0], kbase, HDIM);
  for (int ch = 0; ch < 16; ++ch) {
    stage_tile((uint32_t)(uintptr_t)&ldsV[0][0], vtbase + ch * 64, SEQLEN);
    if (ch + 1 < 16) {
      stage_tile((uint32_t)(uintptr_t)&ldsK[(ch + 1) & 1][0][0],
                 kbase + (size_t)(ch + 1) * 64 * HDIM, HDIM);
      wait_async8();                              // K(ch) and V(ch) landed
    } else {
      wait_async0();
    }
    __syncthreads();
    const float(*kb)[PITCH] = ldsK[ch & 1];
#pragma unroll
    for (int ctl = 0; ctl < 4; ++ctl) {
      const int sl   = ctl * 16 + l16;
      const int scol = ch * 64 + sl;
      v8f c = zero8();
#pragma unroll
      for (int kc = 0; kc < 16; ++kc) {
        const int kbi = kc * 4 + half * 2;
        v2f bf;
        bf[0] = kb[sl][kbi];
        bf[1] = kb[sl][kbi + 1];
        c = wmma_f32(qf[kc], bf, c);
      }
      const float mk = mrow[scol] + (prow[scol] ? -INFINITY : 0.0f);
#pragma unroll
      for (int r = 0; r < 8; ++r) {
        const float p = __expf(c[r] + mk - m_r[r]) * inv_l[r];
        pbase[(size_t)(t0 + r + half * 8) * SEQLEN + scol] = p;  // attn_prob
        pt[wave][r + half * 8][l16] = p;                         // transpose
      }
      asm volatile("s_wait_dscnt 0x0" ::: "memory");  // own-wave LDS visible
#pragma unroll
      for (int kc2 = 0; kc2 < 4; ++kc2) {
        v2f a2;                                   // A: M = lane%16, K = cols
        a2[0] = pt[wave][l16][kc2 * 4 + half * 2];
        a2[1] = pt[wave][l16][kc2 * 4 + half * 2 + 1];
#pragma unroll
        for (int nt = 0; nt < 4; ++nt) {
          const int sv = ctl * 16 + kc2 * 4 + half * 2;  // S offset in chunk
          v2f bf;
          bf[0] = ldsV[nt * 16 + l16][sv];
          bf[1] = ldsV[nt * 16 + l16][sv + 1];
          acc[nt] = wmma_f32(a2, bf, acc[nt]);
        }
      }
    }
    __syncthreads();                              // ldsV / ldsK reusable
  }

  // attn_out laid out (g, t, b, h*64+n) row-major => direct GEMM-A for Wo.
#pragma unroll
  for (int nt = 0; nt < 4; ++nt) {
#pragma unroll
    for (int r = 0; r < 8; ++r) {
      const int t = t0 + r + half * 8;
      attn_out[(((size_t)g * TLEN + t) * BATCH + b) * EMBED + h * HDIM +
               nt * 16 + l16] = acc[nt][r];
    }
  }
}

// ---------------------------------------------------------------------------
// Host-side launch
// ---------------------------------------------------------------------------
extern "C" void kernel_launch(void* const* d_in, const int* in_sizes, int n_in,
                              void* d_out, int out_size, void* d_ws,
                              size_t ws_size, hipStream_t stream) {
  const float* query = (const float*)d_in[0];
  const float* key   = (const float*)d_in[1];
  const unsigned char* kpm = (const unsigned char*)d_in[2];  // bool (B,S)
  const float* gmask = (const float*)d_in[3];
  const float* Wq = (const float*)d_in[4];
  const float* bq = (const float*)d_in[5];
  const float* Wk = (const float*)d_in[6];
  const float* bk = (const float*)d_in[7];
  const float* Wv = (const float*)d_in[8];
  const float* bv = (const float*)d_in[9];
  const float* Wo = (const float*)d_in[10];
  const float* bo = (const float*)d_in[11];

  float* out = (float*)d_out;                            // (G,T,B,D)
  float* probs = out + (size_t)NGRP * TLEN * BATCH * EMBED;  // (BH,G,T,S)

  // Workspace (f32): q 2M | k 2M | v^T 2M | attn_out 8M  => 56 MB
  float* ws = (float*)d_ws;
  float* q_ws  = ws;
  float* k_ws  = q_ws + (size_t)BATCH * NHEAD * TLEN * HDIM;
  float* vt_ws = k_ws + (size_t)BATCH * NHEAD * SEQLEN * HDIM;
  float* ao_ws = vt_ws + (size_t)BATCH * NHEAD * HDIM * SEQLEN;

  const dim3 blk(128);

  // Q/K/V projections (M = 2048 rows each)
  proj_kernel<<<dim3(32, 16), blk, 0, stream>>>(query, Wq, bq, q_ws, 0, SCALE_Q);
  proj_kernel<<<dim3(32, 16), blk, 0, stream>>>(key,   Wk, bk, k_ws, 0, 1.0f);
  proj_kernel<<<dim3(32, 16), blk, 0, stream>>>(key,   Wv, bv, vt_ws, 2, 1.0f);

  // Fused masked-softmax attention: grid (T/64, G, B*H)
  attn_kernel<<<dim3(TLEN / 64, NGRP, BATCH * NHEAD), blk, 0, stream>>>(
      q_ws, k_ws, vt_ws, gmask, kpm, probs, ao_ws);

  // Output projection: M = G*T*B = 8192 rows
  proj_kernel<<<dim3(128, 16), blk, 0, stream>>>(ao_ws, Wo, bo, out, 3, 1.0f);
}